// LAM_Module_37434934952405
// MI455X (gfx1250) — compile-verified
//
#include <hip/hip_runtime.h>
#include <hip/hip_bf16.h>
#include <math.h>

typedef __attribute__((ext_vector_type(2))) float v2f;
typedef __attribute__((ext_vector_type(8))) float v8f;

#define BB 4
#define NN 11
#define DD 589824            // 64*96*96
#define TILE 256
#define TILES (DD / TILE)    // 2304
#define S 128                // slices per batch (128 divides 2304 -> TPS=18)
#define TPS (TILES / S)      // tiles per slice
#define NPAIR 55
#define LDSTRIDE 260         // 11 rows, padded stride (260 % 64 == 4 -> conflict-free WMMA reads)

__device__ __forceinline__ constexpr int pidx(int i, int j) { // requires i<j
    return i * 11 - i * (i + 1) / 2 + (j - i - 1);
}

// ---------------- Kernel 1: per-slice partial energy1 (L1) + energy2 (Gram via WMMA) ---------
__global__ __launch_bounds__(256) void energy_partial_kernel(const float* __restrict__ x,
                                                             float* __restrict__ pe1,
                                                             float* __restrict__ pe2) {
    const int b    = blockIdx.x / S;
    const int s    = blockIdx.x % S;
    const int t    = threadIdx.x;
    const int lane = t & 31;
    const int wave = t >> 5;
    const int lo   = lane & 15;     // N column / A row
    const int hi   = lane >> 4;     // K-pair select

    __shared__ float tile[NN * LDSTRIDE];
    __shared__ float lds1[NPAIR];
    __shared__ float lds2[256];

    if (t < NPAIR) lds1[t] = 0.0f;
    lds2[t] = 0.0f;

    float acc1[NPAIR];
#pragma unroll
    for (int p = 0; p < NPAIR; ++p) acc1[p] = 0.0f;
    v8f c = {};  // WMMA f32 16x16 accumulator (Gram)

    const float* qb      = x + (size_t)b * NN * DD;
    const int    d_begin = s * (TPS * TILE);

    for (int tt = 0; tt < TPS; ++tt) {
        const int d0 = d_begin + tt * TILE;
        // stage 11 x 256 tile into LDS (coalesced per row)
#pragma unroll
        for (int m = 0; m < NN; ++m)
            tile[m * LDSTRIDE + t] = qb[(size_t)m * DD + d0 + t];
        __syncthreads();

        // --- VALU path: pairwise L1, one d-column per thread ---
        float v[NN];
#pragma unroll
        for (int m = 0; m < NN; ++m) v[m] = tile[m * LDSTRIDE + t];
#pragma unroll
        for (int i = 0; i < NN; ++i)
#pragma unroll
            for (int j = i + 1; j < NN; ++j)
                acc1[pidx(i, j)] += fabsf(v[i] - v[j]);

        // --- WMMA path: Gram accumulation, wave owns d-local [wave*32, wave*32+32) ---
        // A-matrix 16x4 f32 layout: lane(0-15)=rows M, VGPR{0,1}=K{0,1}; lanes 16-31 -> K{2,3}.
        // B register layout is the transposed mirror, so feeding identical registers as A and B
        // computes C[M][N] += sum_k q[M][d0+k] * q[N][d0+k]  (rows >= 11 zero-padded).
#pragma unroll
        for (int kk = 0; kk < 8; ++kk) {
            const int dl = wave * 32 + kk * 4 + hi * 2;
            v2f a;
            a.x = (lo < NN) ? tile[lo * LDSTRIDE + dl]     : 0.0f;
            a.y = (lo < NN) ? tile[lo * LDSTRIDE + dl + 1] : 0.0f;
            c = __builtin_amdgcn_wmma_f32_16x16x4_f32(false, a, false, a,
                                                      (short)0, c, false, false);
        }
        __syncthreads();
    }

    // wave-level deterministic reduction of L1 accumulators
#pragma unroll
    for (int p = 0; p < NPAIR; ++p) {
        float vv = acc1[p];
        vv += __shfl_xor(vv, 16, 32);
        vv += __shfl_xor(vv, 8, 32);
        vv += __shfl_xor(vv, 4, 32);
        vv += __shfl_xor(vv, 2, 32);
        vv += __shfl_xor(vv, 1, 32);
        acc1[p] = vv;
    }

    // cross-wave accumulation, serialized for determinism
    for (int w = 0; w < 8; ++w) {
        if (wave == w) {
            if (lane == 0) {
#pragma unroll
                for (int p = 0; p < NPAIR; ++p) lds1[p] += acc1[p];
            }
            // C layout: VGPR r, lanes 0-15 -> (M=r, N=lane); lanes 16-31 -> (M=r+8, N=lane-16)
#pragma unroll
            for (int r = 0; r < 8; ++r)
                lds2[(r + 8 * hi) * 16 + lo] += c[r];
        }
        __syncthreads();
    }

    if (t < NPAIR) pe1[((size_t)b * S + s) * NPAIR + t] = lds1[t];
    if (t < 121) {
        const int i = t / 11, j = t % 11;
        pe2[((size_t)b * S + s) * 121 + t] = lds2[i * 16 + j];
    }
}

// ---------------- Kernel 2: reduce slices, build energy, softmax -> attention ----------------
__global__ __launch_bounds__(128) void attention_kernel(const float* __restrict__ pe1,
                                                        const float* __restrict__ pe2,
                                                        float* __restrict__ att) {
    const int b = blockIdx.x;
    const int t = threadIdx.x;
    __shared__ float en[121];
    __shared__ float ea[121];

    if (t < 121) {
        const int i = t / 11, j = t % 11;
        float e2 = 0.0f;
        for (int s = 0; s < S; ++s) e2 += pe2[((size_t)b * S + s) * 121 + t];
        float e1 = 0.0f;
        if (i != j) {
            const int ii = (i < j) ? i : j;
            const int jj = (i < j) ? j : i;
            const int p  = ii * 11 - ii * (ii + 1) / 2 + (jj - ii - 1);
            for (int s = 0; s < S; ++s) e1 += pe1[((size_t)b * S + s) * NPAIR + p];
        }
        en[t] = e1 * e2;
    }
    __syncthreads();
    if (t < 121) {
        // softmax_j(rowmax - en) == exp(rowmin - en)/sum (rowmax cancels)
        const int i = t / 11;
        float rmin = en[i * 11];
        for (int j = 1; j < 11; ++j) rmin = fminf(rmin, en[i * 11 + j]);
        ea[t] = __expf(rmin - en[t]);
    }
    __syncthreads();
    if (t < 121) {
        const int i = t / 11;
        float ssum = 0.0f;
        for (int j = 0; j < 11; ++j) ssum += ea[i * 11 + j];
        att[b * 121 + t] = ea[t] / ssum;
    }
}

// ---------------- Kernel 3: out = gamma * (att @ q) + x --------------------------------------
__global__ __launch_bounds__(256) void output_kernel(const float* __restrict__ x,
                                                     const float* __restrict__ gamma,
                                                     const float* __restrict__ att,
                                                     float* __restrict__ out) {
    const int b     = blockIdx.x / TILES;
    const int tileI = blockIdx.x % TILES;
    const int t     = threadIdx.x;

    __shared__ float a_s[121];
    if (t < 121) a_s[t] = att[b * 121 + t];
    __syncthreads();

    const float  g    = gamma[0];
    const size_t base = (size_t)b * NN * DD + (size_t)tileI * TILE + t;

    float v[NN];
#pragma unroll
    for (int m = 0; m < NN; ++m) v[m] = x[base + (size_t)m * DD];
#pragma unroll
    for (int n = 0; n < NN; ++n) {
        float acc = 0.0f;
#pragma unroll
        for (int m = 0; m < NN; ++m) acc = fmaf(a_s[n * 11 + m], v[m], acc);
        out[base + (size_t)n * DD] = fmaf(g, acc, v[n]);
    }
}

extern "C" void kernel_launch(void* const* d_in, const int* in_sizes, int n_in,
                              void* d_out, int out_size, void* d_ws, size_t ws_size,
                              hipStream_t stream) {
    const float* x     = (const float*)d_in[0];
    const float* gamma = (const float*)d_in[1];
    float*       out   = (float*)d_out;

    float* ws  = (float*)d_ws;
    float* pe1 = ws;                               // B*S*55
    float* pe2 = pe1 + (size_t)BB * S * NPAIR;     // B*S*121
    float* att = pe2 + (size_t)BB * S * 121;       // B*121

    energy_partial_kernel<<<BB * S, 256, 0, stream>>>(x, pe1, pe2);
    attention_kernel<<<BB, 128, 0, stream>>>(pe1, pe2, att);
    output_kernel<<<BB * TILES, 256, 0, stream>>>(x, gamma, att, out);
}